// AsyncFeatureExtraction_73856257622093
// MI455X (gfx1250) — compile-verified
//
#include <hip/hip_runtime.h>
#include <hip/hip_bf16.h>
#include <math.h>

#define B_    8
#define N_    3072
#define TOUT_ 128
#define CIN_  32
#define DEMB_ 8
#define COUT_ 64
#define BIGF  1.0e10f

typedef __attribute__((ext_vector_type(2))) float v2f;
typedef __attribute__((ext_vector_type(8))) float v8f;

// ---- workspace layout (in floats) ----
// dw   : B*N            = 24576
// hZ   : B*CIN*TOUT     = 32768   (dw-weighted count, cumulative after scan)
// hC   : 32768                     (raw count)
// hS   : 32768                     (sum dw*t)
// hV   : 32768                     (sum dw*v)
// Y    : (B*TOUT) x (CIN*DEMB) = 1024*256 = 262144
#define OFF_DW 0
#define OFF_HZ 24576
#define OFF_HC 57344
#define OFF_HS 90112
#define OFF_HV 122880
#define OFF_Y  155648
#define HIST_TOTAL (4 * 32768)

// ---------------- K0: zero the histogram region ----------------
__global__ void k_zero(float* p, int n) {
    int i = blockIdx.x * blockDim.x + threadIdx.x;
    if (i < n) p[i] = 0.0f;
}

// ---------------- K1: inv_density -> dw (LDS-tiled N^2) ----------------
__global__ __launch_bounds__(256) void k_dw(const float* __restrict__ x,
                                            const float* __restrict__ ks_ptr,
                                            float* __restrict__ dw) {
    const int b   = blockIdx.y;
    const int j   = blockIdx.x * 256 + threadIdx.x;
    const float ks = ks_ptr[0];

    const float tj = x[(size_t)(b * N_ + j) * 3 + 2];
    const int   fj = (int)x[(size_t)(b * N_ + j) * 3 + 0];

    __shared__ float sT[256];
    __shared__ int   sF[256];

    float best = BIGF;
    for (int tile = 0; tile < N_ / 256; ++tile) {
        const int i = tile * 256 + threadIdx.x;
        sT[threadIdx.x] = x[(size_t)(b * N_ + i) * 3 + 2];
        sF[threadIdx.x] = (int)x[(size_t)(b * N_ + i) * 3 + 0];
        __syncthreads();
        #pragma unroll 8
        for (int u = 0; u < 256; ++u) {
            const float d = fabsf(sT[u] - tj);
            // reference: pd = where(same & (pd != 0), pd, BIG); min over i
            if (sF[u] == fj && d != 0.0f) best = fminf(best, d);
        }
        __syncthreads();
    }
    dw[b * N_ + j] = powf(best, ks);   // inv_density ** kernel_scale
}

// ---------------- K2: bucketed accumulation (atomics) ----------------
// inclusion at out-position p iff t_n <= p  <=>  p >= ceil(t_n) (p integer)
__global__ __launch_bounds__(256) void k_hist(const float* __restrict__ x,
                                              const float* __restrict__ dw,
                                              float* __restrict__ hZ,
                                              float* __restrict__ hC,
                                              float* __restrict__ hS,
                                              float* __restrict__ hV) {
    const int idx = blockIdx.x * 256 + threadIdx.x;
    if (idx >= B_ * N_) return;
    const int b = idx / N_;
    const int n = idx - b * N_;

    const float fx = x[(size_t)idx * 3 + 0];
    const float v  = x[(size_t)idx * 3 + 1];
    const float t  = x[(size_t)idx * 3 + 2];

    // padding = any(x != 0)
    if (fx == 0.0f && v == 0.0f && t == 0.0f) return;

    const int c = (int)fx;
    if ((unsigned)c >= (unsigned)CIN_) return;   // onehot vs arange(C) => no bin

    int bucket = (int)ceilf(t);
    if (bucket < 0) bucket = 0;
    if (bucket >= TOUT_) return;                 // never causal

    const float w = dw[b * N_ + n];
    const int base = ((b * CIN_ + c) * TOUT_) + bucket;
    atomicAdd(&hZ[base], w);
    atomicAdd(&hC[base], 1.0f);
    atomicAdd(&hS[base], w * t);
    atomicAdd(&hV[base], w * v);
}

// ---------------- K3: inclusive prefix over the T dimension ----------------
__global__ __launch_bounds__(256) void k_scan(float* __restrict__ hZ,
                                              float* __restrict__ hC,
                                              float* __restrict__ hS,
                                              float* __restrict__ hV) {
    const int lane = blockIdx.x * 256 + threadIdx.x;   // b*CIN + c
    if (lane >= B_ * CIN_) return;
    const int base = lane * TOUT_;
    float az = 0.f, ac = 0.f, as = 0.f, av = 0.f;
    for (int p = 0; p < TOUT_; ++p) {
        az += hZ[base + p]; hZ[base + p] = az;
        ac += hC[base + p]; hC[base + p] = ac;
        as += hS[base + p]; hS[base + p] = as;
        av += hV[base + p]; hV[base + p] = av;
    }
}

// ---------------- K4: assemble fe -> Y (1024 x 256) ----------------
// num[b,t,c,d] = W_dist[d]*(St - pos*Z)/maxpos
//              + (b_dist[d] + emb[c][d] + b_vals[d]) * Z
//              + W_vals[d] * V
// fe = num / (Z+1e-10) / (cnt+1e-10)
__global__ __launch_bounds__(256) void k_fe(const float* __restrict__ outpos,
                                            const float* __restrict__ Wd,
                                            const float* __restrict__ bd,
                                            const float* __restrict__ emb,
                                            const float* __restrict__ Wv,
                                            const float* __restrict__ bv,
                                            const float* __restrict__ hZ,
                                            const float* __restrict__ hC,
                                            const float* __restrict__ hS,
                                            const float* __restrict__ hV,
                                            float* __restrict__ Y) {
    // one block per (b, t); 256 threads cover (c, d)
    const int b = blockIdx.x >> 7;
    const int t = blockIdx.x & (TOUT_ - 1);
    const int c = threadIdx.x >> 3;
    const int d = threadIdx.x & (DEMB_ - 1);

    __shared__ float s_maxpos, s_pos;
    if (threadIdx.x == 0) {
        float mp = outpos[0];
        for (int i = 1; i < TOUT_; ++i) mp = fmaxf(mp, outpos[i]);
        s_maxpos = mp;
        s_pos = outpos[t];
    }
    __syncthreads();
    const float maxpos = s_maxpos;
    const float pos    = s_pos;

    const int base = ((b * CIN_ + c) * TOUT_) + t;
    const float Z  = hZ[base];
    const float Cn = hC[base];
    const float St = hS[base];
    const float V  = hV[base];

    const float num = Wd[d] * ((St - pos * Z) / maxpos)
                    + (bd[d] + emb[c * DEMB_ + d] + bv[d]) * Z
                    + Wv[d] * V;
    const float fe = num / (Z + 1e-10f) / (Cn + 1e-10f);

    Y[(size_t)(b * TOUT_ + t) * (CIN_ * DEMB_) + (c * DEMB_ + d)] = fe;
}

// ---------------- K5: WMMA f32 GEMM (1024x256)x(256x64) + bias, ----------------
// store transposed out[b, o, t]
__global__ __launch_bounds__(128) void k_gemm(const float* __restrict__ Y,
                                              const float* __restrict__ Wl,
                                              const float* __restrict__ bl,
                                              float* __restrict__ out) {
    const int lane = threadIdx.x & 31;
    const int wave = threadIdx.x >> 5;
    const int tile = blockIdx.x * 4 + wave;   // 256 tiles
    const int tile_n = tile & 3;              // 4 tiles over 64 outputs
    const int tile_m = tile >> 2;             // 64 tiles over 1024 rows
    const int half = lane >> 4;
    const int r    = lane & 15;

    // A: row M = tile_m*16 + r of Y; lane half selects K pair {0,1} / {2,3}
    const float* arow = Y  + (size_t)(tile_m * 16 + r) * 256;
    // B[k][n] = W_lin[n][k]; lane r = column n, half selects K pair
    const float* brow = Wl + (size_t)(tile_n * 16 + r) * 256;

    v8f acc = {};
    #pragma unroll 4
    for (int k = 0; k < 256; k += 4) {
        v2f a = *(const v2f*)(arow + k + 2 * half);
        v2f b = *(const v2f*)(brow + k + 2 * half);
        acc = __builtin_amdgcn_wmma_f32_16x16x4_f32(
            /*neg_a=*/false, a, /*neg_b=*/false, b,
            /*c_mod=*/(short)0, acc, /*reuse_a=*/false, /*reuse_b=*/false);
    }

    const int o = tile_n * 16 + r;
    const float bias = bl[o];
    #pragma unroll
    for (int g = 0; g < 8; ++g) {
        const int m  = tile_m * 16 + g + half * 8;  // global row = b*128 + t
        const int bb = m >> 7;
        const int tt = m & (TOUT_ - 1);
        out[(size_t)(bb * COUT_ + o) * TOUT_ + tt] = acc[g] + bias;
    }
}

extern "C" void kernel_launch(void* const* d_in, const int* in_sizes, int n_in,
                              void* d_out, int out_size, void* d_ws, size_t ws_size,
                              hipStream_t stream) {
    const float* x       = (const float*)d_in[0];
    const float* outpos  = (const float*)d_in[1];
    const float* W_dist  = (const float*)d_in[2];
    const float* b_dist  = (const float*)d_in[3];
    const float* emb     = (const float*)d_in[4];
    const float* W_vals  = (const float*)d_in[5];
    const float* b_vals  = (const float*)d_in[6];
    const float* W_lin   = (const float*)d_in[7];
    const float* b_lin   = (const float*)d_in[8];
    const float* kscale  = (const float*)d_in[9];
    float* out = (float*)d_out;

    float* ws = (float*)d_ws;
    float* dw = ws + OFF_DW;
    float* hZ = ws + OFF_HZ;
    float* hC = ws + OFF_HC;
    float* hS = ws + OFF_HS;
    float* hV = ws + OFF_HV;
    float* Y  = ws + OFF_Y;

    // K0: zero histograms
    k_zero<<<(HIST_TOTAL + 255) / 256, 256, 0, stream>>>(hZ, HIST_TOTAL);
    // K1: dw = inv_density ** kernel_scale
    k_dw<<<dim3(N_ / 256, B_), 256, 0, stream>>>(x, kscale, dw);
    // K2: bucketed atomics
    k_hist<<<(B_ * N_ + 255) / 256, 256, 0, stream>>>(x, dw, hZ, hC, hS, hV);
    // K3: prefix scan over T
    k_scan<<<1, 256, 0, stream>>>(hZ, hC, hS, hV);
    // K4: assemble fe rows
    k_fe<<<B_ * TOUT_, 256, 0, stream>>>(outpos, W_dist, b_dist, emb, W_vals,
                                         b_vals, hZ, hC, hS, hV, Y);
    // K5: WMMA GEMM + bias + transpose
    k_gemm<<<64, 128, 0, stream>>>(Y, W_lin, b_lin, out);
}